// DeeperGCN_13907104104744
// MI455X (gfx1250) — compile-verified
//
#include <hip/hip_runtime.h>
#include <hip/hip_bf16.h>
#include <math.h>

#define N_NODES 50000
#define N_EDGES 800000
#define HDIM    64
#define H2      128
#define NLAYERS 3
#define C_OUT   4

typedef __attribute__((ext_vector_type(2))) float v2f;
typedef __attribute__((ext_vector_type(8))) float v8f;

// ---- monotone float<->uint mapping for atomic segment-max on f32 ----
__device__ __forceinline__ unsigned fmono(float x) {
  unsigned u = __float_as_uint(x);
  return (u & 0x80000000u) ? ~u : (u | 0x80000000u);
}
__device__ __forceinline__ float fmono_dec(unsigned m) {
  return (m & 0x80000000u) ? __uint_as_float(m ^ 0x80000000u) : __uint_as_float(~m);
}
#define MX_INIT 0x007FFFFFu /* fmono(-inf) */

// wave-uniform value -> SGPR (tid>>5 is uniform per wave but LLVM can't prove it)
__device__ __forceinline__ int uniform_wave_id() {
  return __builtin_amdgcn_readfirstlane(
      (int)((blockIdx.x * blockDim.x + threadIdx.x) >> 5));
}

// =====================================================================
// f32 WMMA GEMM, compile-time K and NCOLS:
//   C[M,NCOLS] = A[M,K] @ B[K,NCOLS] + bias (+ res)
// One wave per 16-row strip. A-fragments (K/4 x v2f) preloaded once and
// reused across all NCOLS/16 output tiles; k-loop and tile-loop fully
// unrolled so every global_load gets an immediate offset (no per-step
// 64-bit address math between WMMAs).
// =====================================================================
template <int K, int NCOLS, bool RES>
__global__ void gemm_wmma(const float* __restrict__ A, const float* __restrict__ B,
                          const float* __restrict__ bias, const float* __restrict__ res,
                          float* __restrict__ C, int M) {
  constexpr int KT = K / 4;
  constexpr int NT = NCOLS / 16;
  const int strip = uniform_wave_id();          // 16-row strip index (SGPR)
  const int lane  = threadIdx.x & 31;
  if (strip * 16 >= M) return;                  // wave-uniform exit; EXEC all-1s inside
  const int hi  = lane >> 4;                    // lanes 0-15 vs 16-31 (K-pair select)
  const int l16 = lane & 15;

  // Preload A fragments for this strip: lane l16 holds row strip*16+l16,
  // half hi holds K = 4k+2*hi, 4k+2*hi+1.
  v2f afrag[KT];
  const float* ap = A + (size_t)(strip * 16 + l16) * K + hi * 2;
#pragma unroll
  for (int kk = 0; kk < KT; ++kk) {
    afrag[kk][0] = ap[kk * 4];
    afrag[kk][1] = ap[kk * 4 + 1];
  }

#pragma unroll
  for (int nt = 0; nt < NT; ++nt) {
    const int col = nt * 16 + l16;
    const float* bp = B + (size_t)(hi * 2) * NCOLS + col;
    v8f acc = {0.f, 0.f, 0.f, 0.f, 0.f, 0.f, 0.f, 0.f};
#pragma unroll
    for (int kk = 0; kk < KT; ++kk) {
      v2f b;
      b[0] = bp[(kk * 4)     * NCOLS];
      b[1] = bp[(kk * 4 + 1) * NCOLS];
      acc = __builtin_amdgcn_wmma_f32_16x16x4_f32(false, afrag[kk], false, b,
                                                  (short)0, acc, false, false);
    }
    // C layout: VGPR g holds row strip*16 + hi*8 + g, col
    const float bb = bias[col];
    float* cp = C + (size_t)(strip * 16 + hi * 8) * NCOLS + col;
    const float* rp = RES ? (res + (size_t)(strip * 16 + hi * 8) * NCOLS + col) : nullptr;
#pragma unroll
    for (int g = 0; g < 8; ++g) {
      float v = acc[g] + bb;
      if (RES) v += rp[(size_t)g * NCOLS];
      cp[(size_t)g * NCOLS] = v;
    }
  }
}

// =====================================================================
// Edge pass 1: scaled = (relu(z[src]+ea)+eps)*t ; atomic segment-max into mx
// One wave per edge (index scalarized), 2 features per lane.
// =====================================================================
__global__ void edge_softmax_max(const float* __restrict__ z, const float* __restrict__ ea,
                                 const long long* __restrict__ ei,
                                 const float* __restrict__ t_all, int layer,
                                 unsigned* __restrict__ mx) {
  const int e    = uniform_wave_id();
  const int lane = threadIdx.x & 31;
  if (e >= N_EDGES) return;
  __builtin_prefetch(ea + ((long)e + 64) * HDIM, 0, 0);   // stream-ahead on edge feats
  const float t = t_all[layer];
  const long long s = ei[e];                               // scalar loads (uniform addr)
  const long long d = ei[(long)N_EDGES + e];
  const int f = lane * 2;
  const float2 zr = *(const float2*)(z + (long)s * HDIM + f);
  const float2 er = *(const float2*)(ea + (long)e * HDIM + f);
  const float m0 = fmaxf(zr.x + er.x, 0.0f) + 1e-7f;
  const float m1 = fmaxf(zr.y + er.y, 0.0f) + 1e-7f;
  unsigned* p = mx + (long)d * HDIM + f;
  atomicMax(p,     fmono(m0 * t));
  atomicMax(p + 1, fmono(m1 * t));
}

// Decode mx back to float; empty segments (-inf) -> 0 (matches isfinite guard)
__global__ void mx_finalize(unsigned* __restrict__ mx, int n) {
  const int i = blockIdx.x * blockDim.x + threadIdx.x;
  if (i >= n) return;
  float v = fmono_dec(mx[i]);
  if (!__builtin_isfinite(v)) v = 0.0f;
  ((float*)mx)[i] = v;
}

// =====================================================================
// Edge pass 2: ex = exp(scaled - mx[dst]); den += ex; num += ex*msg
// (agg = num/den later -- alpha never materialized)
// =====================================================================
__global__ void edge_softmax_accum(const float* __restrict__ z, const float* __restrict__ ea,
                                   const long long* __restrict__ ei,
                                   const float* __restrict__ t_all, int layer,
                                   const float* __restrict__ mxf,
                                   float* __restrict__ den, float* __restrict__ num) {
  const int e    = uniform_wave_id();
  const int lane = threadIdx.x & 31;
  if (e >= N_EDGES) return;
  __builtin_prefetch(ea + ((long)e + 64) * HDIM, 0, 0);
  const float t = t_all[layer];
  const long long s = ei[e];
  const long long d = ei[(long)N_EDGES + e];
  const int f = lane * 2;
  const float2 zr = *(const float2*)(z + (long)s * HDIM + f);
  const float2 er = *(const float2*)(ea + (long)e * HDIM + f);
  const float m0 = fmaxf(zr.x + er.x, 0.0f) + 1e-7f;
  const float m1 = fmaxf(zr.y + er.y, 0.0f) + 1e-7f;
  const float2 mxv = *(const float2*)(mxf + (long)d * HDIM + f);
  const float e0 = __expf(m0 * t - mxv.x);
  const float e1 = __expf(m1 * t - mxv.y);
  float* dp = den + (long)d * HDIM + f;
  float* np = num + (long)d * HDIM + f;
  atomicAdd(dp,     e0);
  atomicAdd(dp + 1, e1);
  atomicAdd(np,     e0 * m0);
  atomicAdd(np + 1, e1 * m1);
}

// Initialize softmax scratch (d_ws is poisoned by harness; must re-init every call)
__global__ void softmax_init(unsigned* __restrict__ mx, float* __restrict__ den,
                             float* __restrict__ num, int n) {
  const int i = blockIdx.x * blockDim.x + threadIdx.x;
  if (i >= n) return;
  mx[i] = MX_INIT; den[i] = 0.0f; num[i] = 0.0f;
}

// hin = num/(den+1e-16) + xin   (root skip)
__global__ void agg_combine(const float* __restrict__ num, const float* __restrict__ den,
                            const float* __restrict__ xin, float* __restrict__ hin, int n) {
  const int i = blockIdx.x * blockDim.x + threadIdx.x;
  if (i >= n) return;
  hin[i] = num[i] / (den[i] + 1e-16f) + xin[i];
}

// =====================================================================
// z = relu(LayerNorm64(h)) -- one wave per row, wave32 shuffle reductions
// =====================================================================
__global__ void prenorm_relu(const float* __restrict__ h, const float* __restrict__ g,
                             const float* __restrict__ b, float* __restrict__ z) {
  const int n    = uniform_wave_id();
  const int lane = threadIdx.x & 31;
  if (n >= N_NODES) return;
  const float2 v = *(const float2*)(h + (long)n * HDIM + lane * 2);
  float s = v.x + v.y;
#pragma unroll
  for (int o = 16; o >= 1; o >>= 1) s += __shfl_xor(s, o, 32);
  const float mu = s * (1.0f / 64.0f);
  const float dx = v.x - mu, dy = v.y - mu;
  float q = dx * dx + dy * dy;
#pragma unroll
  for (int o = 16; o >= 1; o >>= 1) q += __shfl_xor(q, o, 32);
  const float rs = rsqrtf(q * (1.0f / 64.0f) + 1e-5f);
  float2 o2;
  o2.x = fmaxf(dx * rs * g[lane * 2]     + b[lane * 2],     0.0f);
  o2.y = fmaxf(dy * rs * g[lane * 2 + 1] + b[lane * 2 + 1], 0.0f);
  *(float2*)(z + (long)n * HDIM + lane * 2) = o2;
}

// In-place relu(LayerNorm128(t1)) for the MLP hidden -- 4 feats/lane
__global__ void mlp_ln_relu(float* __restrict__ t1, const float* __restrict__ g,
                            const float* __restrict__ b) {
  const int n    = uniform_wave_id();
  const int lane = threadIdx.x & 31;
  if (n >= N_NODES) return;
  float4 v = *(float4*)(t1 + (long)n * H2 + lane * 4);
  float s = v.x + v.y + v.z + v.w;
#pragma unroll
  for (int o = 16; o >= 1; o >>= 1) s += __shfl_xor(s, o, 32);
  const float mu = s * (1.0f / 128.0f);
  const float d0 = v.x - mu, d1 = v.y - mu, d2 = v.z - mu, d3 = v.w - mu;
  float q = d0 * d0 + d1 * d1 + d2 * d2 + d3 * d3;
#pragma unroll
  for (int o = 16; o >= 1; o >>= 1) q += __shfl_xor(q, o, 32);
  const float rs = rsqrtf(q * (1.0f / 128.0f) + 1e-5f);
  const float4 g4 = *(const float4*)(g + lane * 4);
  const float4 b4 = *(const float4*)(b + lane * 4);
  float4 o4;
  o4.x = fmaxf(d0 * rs * g4.x + b4.x, 0.0f);
  o4.y = fmaxf(d1 * rs * g4.y + b4.y, 0.0f);
  o4.z = fmaxf(d2 * rs * g4.z + b4.z, 0.0f);
  o4.w = fmaxf(d3 * rs * g4.w + b4.w, 0.0f);
  *(float4*)(t1 + (long)n * H2 + lane * 4) = o4;
}

// =====================================================================
// Head: out = relu(LN64(h; ln_g0, ln_b0)) @ lin_w + lin_b  (one wave/row)
// =====================================================================
__global__ void head_kernel(const float* __restrict__ h, const float* __restrict__ g,
                            const float* __restrict__ b, const float* __restrict__ W,
                            const float* __restrict__ lb, float* __restrict__ out) {
  const int n    = uniform_wave_id();
  const int lane = threadIdx.x & 31;
  if (n >= N_NODES) return;
  const float2 v = *(const float2*)(h + (long)n * HDIM + lane * 2);
  float s = v.x + v.y;
#pragma unroll
  for (int o = 16; o >= 1; o >>= 1) s += __shfl_xor(s, o, 32);
  const float mu = s * (1.0f / 64.0f);
  const float dx = v.x - mu, dy = v.y - mu;
  float q = dx * dx + dy * dy;
#pragma unroll
  for (int o = 16; o >= 1; o >>= 1) q += __shfl_xor(q, o, 32);
  const float rs = rsqrtf(q * (1.0f / 64.0f) + 1e-5f);
  const float f0 = fmaxf(dx * rs * g[lane * 2]     + b[lane * 2],     0.0f);
  const float f1 = fmaxf(dy * rs * g[lane * 2 + 1] + b[lane * 2 + 1], 0.0f);
  const float4 w0 = *(const float4*)(W + (lane * 2) * C_OUT);
  const float4 w1 = *(const float4*)(W + (lane * 2 + 1) * C_OUT);
  float4 p;
  p.x = f0 * w0.x + f1 * w1.x;
  p.y = f0 * w0.y + f1 * w1.y;
  p.z = f0 * w0.z + f1 * w1.z;
  p.w = f0 * w0.w + f1 * w1.w;
#pragma unroll
  for (int o = 16; o >= 1; o >>= 1) {
    p.x += __shfl_xor(p.x, o, 32);
    p.y += __shfl_xor(p.y, o, 32);
    p.z += __shfl_xor(p.z, o, 32);
    p.w += __shfl_xor(p.w, o, 32);
  }
  if (lane == 0) {
    float* op = out + (long)n * C_OUT;
    op[0] = p.x + lb[0]; op[1] = p.y + lb[1];
    op[2] = p.z + lb[2]; op[3] = p.w + lb[3];
  }
}

// =====================================================================
extern "C" void kernel_launch(void* const* d_in, const int* in_sizes, int n_in,
                              void* d_out, int out_size, void* d_ws, size_t ws_size,
                              hipStream_t stream) {
  (void)in_sizes; (void)n_in; (void)out_size; (void)ws_size;
  const float*     x      = (const float*)d_in[0];
  const float*     eattr  = (const float*)d_in[1];
  const long long* ei     = (const long long*)d_in[2];   // int64 per reference
  const float*     enc_w  = (const float*)d_in[3];
  const float*     enc_b  = (const float*)d_in[4];
  const float*     eenc_w = (const float*)d_in[5];
  const float*     eenc_b = (const float*)d_in[6];
  const float*     mlp_w1 = (const float*)d_in[7];
  const float*     mlp_b1 = (const float*)d_in[8];
  const float*     mlp_lng= (const float*)d_in[9];
  const float*     mlp_lnb= (const float*)d_in[10];
  const float*     mlp_w2 = (const float*)d_in[11];
  const float*     mlp_b2 = (const float*)d_in[12];
  const float*     t_all  = (const float*)d_in[13];
  const float*     ln_g   = (const float*)d_in[14];
  const float*     ln_b   = (const float*)d_in[15];
  const float*     lin_w  = (const float*)d_in[16];
  const float*     lin_b  = (const float*)d_in[17];
  float* out = (float*)d_out;

  // Workspace layout (re-initialized each call where needed)
  float* ws = (float*)d_ws;
  float*    h   = ws; ws += (size_t)N_NODES * HDIM;
  float*    z   = ws; ws += (size_t)N_NODES * HDIM;
  float*    ea  = ws; ws += (size_t)N_EDGES * HDIM;
  unsigned* mx  = (unsigned*)ws; ws += (size_t)N_NODES * HDIM;
  float*    den = ws; ws += (size_t)N_NODES * HDIM;
  float*    num = ws; ws += (size_t)N_NODES * HDIM;
  float*    hin = ws; ws += (size_t)N_NODES * HDIM;
  float*    t1  = ws; ws += (size_t)N_NODES * H2;

  const int TB = 256;                               // 8 waves per block
  auto strip_blocks = [](long M) { return (int)((M / 16 + 7) / 8); };
  const int nodeWaveBlocks = (N_NODES + 7) / 8;     // one wave per node row
  const int edgeWaveBlocks = (N_EDGES + 7) / 8;     // one wave per edge
  const int nfElems = N_NODES * HDIM;
  const int nfBlocks = (nfElems + TB - 1) / TB;

  // Encoders
  gemm_wmma<64, HDIM, false><<<strip_blocks(N_NODES), TB, 0, stream>>>(
      x, enc_w, enc_b, nullptr, h, N_NODES);
  gemm_wmma<16, HDIM, false><<<strip_blocks(N_EDGES), TB, 0, stream>>>(
      eattr, eenc_w, eenc_b, nullptr, ea, N_EDGES);

  for (int l = 0; l < NLAYERS; ++l) {
    const float* convin = (l == 0) ? h : z;
    if (l > 0)
      prenorm_relu<<<nodeWaveBlocks, TB, 0, stream>>>(
          h, ln_g + (size_t)l * HDIM, ln_b + (size_t)l * HDIM, z);

    softmax_init<<<nfBlocks, TB, 0, stream>>>(mx, den, num, nfElems);
    edge_softmax_max<<<edgeWaveBlocks, TB, 0, stream>>>(convin, ea, ei, t_all, l, mx);
    mx_finalize<<<nfBlocks, TB, 0, stream>>>(mx, nfElems);
    edge_softmax_accum<<<edgeWaveBlocks, TB, 0, stream>>>(
        convin, ea, ei, t_all, l, (const float*)mx, den, num);
    agg_combine<<<nfBlocks, TB, 0, stream>>>(num, den, convin, hin, nfElems);

    // MLP: t1 = hin @ w1 + b1 ; relu(LN128) in-place ; h = [h +] t1 @ w2 + b2
    gemm_wmma<64, H2, false><<<strip_blocks(N_NODES), TB, 0, stream>>>(
        hin, mlp_w1 + (size_t)l * HDIM * H2, mlp_b1 + (size_t)l * H2, nullptr,
        t1, N_NODES);
    mlp_ln_relu<<<nodeWaveBlocks, TB, 0, stream>>>(
        t1, mlp_lng + (size_t)l * H2, mlp_lnb + (size_t)l * H2);
    if (l == 0)
      gemm_wmma<128, HDIM, false><<<strip_blocks(N_NODES), TB, 0, stream>>>(
          t1, mlp_w2, mlp_b2, nullptr, h, N_NODES);
    else
      gemm_wmma<128, HDIM, true><<<strip_blocks(N_NODES), TB, 0, stream>>>(
          t1, mlp_w2 + (size_t)l * H2 * HDIM, mlp_b2 + (size_t)l * HDIM,
          h, h, N_NODES);
  }

  // Head uses layer-0's norm params
  head_kernel<<<nodeWaveBlocks, TB, 0, stream>>>(h, ln_g, ln_b, lin_w, lin_b, out);
}